// HCHA_61538291417104
// MI455X (gfx1250) — compile-verified
//
#include <hip/hip_runtime.h>
#include <math.h>

typedef __attribute__((ext_vector_type(2))) float v2f;
typedef __attribute__((ext_vector_type(8))) float v8f;

#define N_NODES 100000
#define N_EDGES 50000
#define N_INC   600000
#define F       128

// ---------------------------------------------------------------------------
// GEMM: Y[M,128] = X[M,128] @ W[128,128]   (fp32 WMMA 16x16x4, exact fp32)
// Block = 256 threads = 8 waves; each wave computes a 16x128 output strip.
// W is staged in LDS pair-interleaved as lds[k/2][n][k%2] so a B fragment
// (W[ka][n], W[ka+1][n]) is one contiguous 8-byte ds_load_b64 into an even-
// aligned VGPR pair (no re-pairing movs). K loop fully unrolled so A global
// loads can be hoisted/pipelined by the scheduler.
// M % 16 == 0, so strip validity is wave-uniform and EXEC is all-ones around
// every WMMA (ISA requirement).
// ---------------------------------------------------------------------------
__global__ __launch_bounds__(256) void gemm128_wmma(
    const float* __restrict__ X, const float* __restrict__ W,
    float* __restrict__ Y, int M)
{
  __shared__ __align__(16) float ldsW[128 * 128];   // [64 pairs][128 n][2]
  const int tid = threadIdx.x;
  // cooperative load of W: 4096 float4 reads, scattered into interleaved layout
  for (int i = tid; i < 128 * 128 / 4; i += 256) {
    float4 w = ((const float4*)W)[i];
    int k  = i >> 5;           // source row (0..127)
    int n0 = (i & 31) * 4;     // source col
    int base = (k >> 1) * 256 + (k & 1);
    ldsW[base + (n0 + 0) * 2] = w.x;
    ldsW[base + (n0 + 1) * 2] = w.y;
    ldsW[base + (n0 + 2) * 2] = w.z;
    ldsW[base + (n0 + 3) * 2] = w.w;
  }
  __syncthreads();

  const int wave = tid >> 5;
  const int lane = tid & 31;
  const int lhi  = lane >> 4;   // 0: lanes 0-15, 1: lanes 16-31
  const int llo  = lane & 15;
  const long row0 = (long)blockIdx.x * 128 + wave * 16;
  if (row0 >= M) return;        // wave-uniform (M multiple of 16)

  const float* arow = X + (row0 + llo) * (long)F;

  v8f acc[8];
#pragma unroll
  for (int nt = 0; nt < 8; ++nt) acc[nt] = (v8f)0.0f;

#pragma unroll
  for (int k0 = 0; k0 < 128; k0 += 4) {
    const int ka = k0 + 2 * lhi;             // A: v0 holds K0/K2 halves, v1 K1/K3
    const v2f a = *(const v2f*)(arow + ka);  // contiguous 8B global load
    const int pb = (ka >> 1) * 256;          // pair row in interleaved LDS
#pragma unroll
    for (int nt = 0; nt < 8; ++nt) {
      const v2f b = *(const v2f*)(ldsW + pb + (nt * 16 + llo) * 2);  // ds_load_b64
      acc[nt] = __builtin_amdgcn_wmma_f32_16x16x4_f32(
          false, a, false, b, (short)0, acc[nt], false, false);
    }
  }

  // C/D layout: VGPR v -> M = v + 8*lhi, N = llo (per 16x16 tile)
#pragma unroll
  for (int nt = 0; nt < 8; ++nt) {
#pragma unroll
    for (int v = 0; v < 8; ++v) {
      Y[(row0 + 8 * lhi + v) * (long)F + nt * 16 + llo] = acc[nt][v];
    }
  }
}

// ---------------------------------------------------------------------------
// Graph-structure degree counting (done once): D[node]+=1, B[edge]+=1
// ---------------------------------------------------------------------------
__global__ __launch_bounds__(256) void degree_count(
    const int* __restrict__ nidx, const int* __restrict__ eidx,
    float* __restrict__ D, float* __restrict__ B, int n)
{
  int i = blockIdx.x * 256 + threadIdx.x;
  if (i >= n) return;
  atomicAdd(&D[nidx[i]], 1.0f);
  atomicAdd(&B[eidx[i]], 1.0f);
}

__global__ __launch_bounds__(256) void invert_deg(float* __restrict__ a, int n)
{
  int i = blockIdx.x * 256 + threadIdx.x;
  if (i >= n) return;
  float v = a[i];
  a[i] = (v > 0.0f) ? (1.0f / v) : 0.0f;
}

// ---------------------------------------------------------------------------
// Segment scatter: dst[tidx[i]] += scale[tidx[i]] * src[sidx[i]]  (128 feats)
// One thread per (incidence, 4 features): float4 load + 4 f32 atomics.
// ---------------------------------------------------------------------------
__global__ __launch_bounds__(256) void scatter_fma4(
    const float* __restrict__ src, const float* __restrict__ scale,
    const int* __restrict__ sidx, const int* __restrict__ tidx,
    float* __restrict__ dst)
{
  long i   = (long)blockIdx.x * 256 + threadIdx.x;
  long inc = i >> 5;
  if (inc >= N_INC) return;
  int  f = (int)(i & 31) * 4;
  long s = sidx[inc];
  long t = tidx[inc];
  float sc = scale[t];
  const float4 v = *(const float4*)(src + s * (long)F + f);
  float* dp = dst + t * (long)F + f;
  atomicAdd(dp + 0, sc * v.x);
  atomicAdd(dp + 1, sc * v.y);
  atomicAdd(dp + 2, sc * v.z);
  atomicAdd(dp + 3, sc * v.w);
}

// ---------------------------------------------------------------------------
// y = y + b  (optionally followed by ELU: x>0 ? x : exp(x)-1)
// ---------------------------------------------------------------------------
__global__ __launch_bounds__(256) void bias_act(
    float* __restrict__ y, const float* __restrict__ b, long total, int do_elu)
{
  long i = (long)blockIdx.x * 256 + threadIdx.x;
  if (i >= total) return;
  float v = y[i] + b[i & (F - 1)];
  if (do_elu) v = (v > 0.0f) ? v : expm1f(v);
  y[i] = v;
}

// ---------------------------------------------------------------------------
// Host-side orchestration
// ---------------------------------------------------------------------------
static inline void run_layer(const float* in, const float* W, const float* b,
                             float* Ybuf, float* efeat, float* outbuf,
                             const float* Dinv, const float* Binv,
                             const int* nidx, const int* eidx,
                             int do_elu, hipStream_t stream)
{
  const long node_elems = (long)N_NODES * F;
  const long edge_elems = (long)N_EDGES * F;
  const int  gemm_blocks    = (N_NODES + 127) / 128;
  const int  scatter_blocks = (N_INC * 32) / 256;
  const int  bias_blocks    = (int)((node_elems + 255) / 256);

  gemm128_wmma<<<gemm_blocks, 256, 0, stream>>>(in, W, Ybuf, N_NODES);

  hipMemsetAsync(efeat, 0, edge_elems * sizeof(float), stream);
  // node -> hyperedge: e_feat[e] += Binv[e] * Y[n]
  scatter_fma4<<<scatter_blocks, 256, 0, stream>>>(Ybuf, Binv, nidx, eidx, efeat);

  hipMemsetAsync(outbuf, 0, node_elems * sizeof(float), stream);
  // hyperedge -> node: out[n] += Dinv[n] * e_feat[e]
  scatter_fma4<<<scatter_blocks, 256, 0, stream>>>(efeat, Dinv, eidx, nidx, outbuf);

  bias_act<<<bias_blocks, 256, 0, stream>>>(outbuf, b, node_elems, do_elu);
}

extern "C" void kernel_launch(void* const* d_in, const int* in_sizes, int n_in,
                              void* d_out, int out_size, void* d_ws, size_t ws_size,
                              hipStream_t stream)
{
  const float* x     = (const float*)d_in[0];
  const int*   edges = (const int*)d_in[1];          // [2, N_INC] int32
  const int*   nidx  = edges;
  const int*   eidx  = edges + N_INC;
  // d_in[2] = edge_weight (unused by the reference model)
  const float* W1 = (const float*)d_in[3];
  const float* b1 = (const float*)d_in[4];
  const float* W2 = (const float*)d_in[5];
  const float* b2 = (const float*)d_in[6];
  const float* W3 = (const float*)d_in[7];
  const float* b3 = (const float*)d_in[8];
  float* out = (float*)d_out;

  // Workspace layout (floats):
  float* ws    = (float*)d_ws;
  float* nodeA = ws;                                  // 100000*128
  float* nodeB = nodeA + (long)N_NODES * F;           // 100000*128
  float* efeat = nodeB + (long)N_NODES * F;           //  50000*128
  float* Dinv  = efeat + (long)N_EDGES * F;           // 100000
  float* Binv  = Dinv + N_NODES;                      //  50000

  // ---- degrees (graph-structure only; computed once) ----
  hipMemsetAsync(Dinv, 0, (size_t)(N_NODES + N_EDGES) * sizeof(float), stream);
  degree_count<<<(N_INC + 255) / 256, 256, 0, stream>>>(nidx, eidx, Dinv, Binv, N_INC);
  invert_deg<<<(N_NODES + 255) / 256, 256, 0, stream>>>(Dinv, N_NODES);
  invert_deg<<<(N_EDGES + 255) / 256, 256, 0, stream>>>(Binv, N_EDGES);

  // ---- layer 1: in = x,      Y = nodeA, out = nodeB, ELU ----
  run_layer(x,     W1, b1, nodeA, efeat, nodeB, Dinv, Binv, nidx, eidx, 1, stream);
  // ---- layer 2: in = nodeB,  Y = nodeA, out = nodeA (alias ok), ELU ----
  run_layer(nodeB, W2, b2, nodeA, efeat, nodeA, Dinv, Binv, nidx, eidx, 1, stream);
  // ---- layer 3: in = nodeA,  Y = nodeB, out = d_out, no ELU ----
  run_layer(nodeA, W3, b3, nodeB, efeat, out,   Dinv, Binv, nidx, eidx, 0, stream);
}